// KalmanFilter_16673063043646
// MI455X (gfx1250) — compile-verified
//
#include <hip/hip_runtime.h>
#include <math.h>

#define B_   128
#define T_   100
#define DZ_  64
#define DX_  128
#define NTHR 128   // 4 wave32s per workgroup

typedef __attribute__((ext_vector_type(2))) float v2f;
typedef __attribute__((ext_vector_type(8))) float v8f;

// Workgroup-cooperative GEMM on LDS-resident f32 matrices using
// V_WMMA_F32_16X16X4_F32.  D = op(A)*op(B) (+ C if ACC).
// M,N multiples of 16; K multiple of 4.  Uniform control flow across the WG.
template<int M, int N, int K, bool TA, bool TB, bool ACC>
__device__ __forceinline__ void wg_gemm(const float* __restrict__ Am, int lda,
                                        const float* __restrict__ Bm, int ldb,
                                        float* __restrict__ Dm, int ldd,
                                        const float* __restrict__ Cm, int ldc,
                                        int tid)
{
    const int lane = tid & 31;
    const int wave = tid >> 5;
    const int r = lane & 15;   // row (A) / col (B,D) within tile
    const int h = lane >> 4;   // half-wave selector
    constexpr int MT = M / 16, NT = N / 16, KT = K / 4;
    for (int tile = wave; tile < MT * NT; tile += NTHR / 32) {
        const int mt = tile / NT, nt = tile % NT;
        const int m0 = mt * 16, n0 = nt * 16;
        v8f acc = {};
        if constexpr (ACC) {
#pragma unroll
            for (int v = 0; v < 8; ++v)
                acc[v] = Cm[(m0 + v + 8 * h) * ldc + n0 + r];
        }
#pragma unroll 4
        for (int kt = 0; kt < KT; ++kt) {
            const int k0 = kt * 4 + 2 * h;
            v2f af, bf;
            if constexpr (TA) {   // A fragment: 16x4, lane = M, vgpr v -> K = k0+v
                af.x = Am[(k0    ) * lda + m0 + r];
                af.y = Am[(k0 + 1) * lda + m0 + r];
            } else {
                af.x = Am[(m0 + r) * lda + k0];
                af.y = Am[(m0 + r) * lda + k0 + 1];
            }
            if constexpr (TB) {   // B fragment: 4x16, lane = N, vgpr v -> K = k0+v
                bf.x = Bm[(n0 + r) * ldb + k0];
                bf.y = Bm[(n0 + r) * ldb + k0 + 1];
            } else {
                bf.x = Bm[(k0    ) * ldb + n0 + r];
                bf.y = Bm[(k0 + 1) * ldb + n0 + r];
            }
            acc = __builtin_amdgcn_wmma_f32_16x16x4_f32(false, af, false, bf,
                                                        (short)0, acc, false, false);
        }
#pragma unroll
        for (int v = 0; v < 8; ++v)     // D tile: vgpr v -> M = v + 8h
            Dm[(m0 + v + 8 * h) * ldd + n0 + r] = acc[v];
    }
}

// In-place lower Cholesky of SPD n x n matrix in LDS (lower triangle + diag
// overwritten; strict upper triangle left untouched).
__device__ __forceinline__ void wg_cholesky(float* Sm, int n, int ld, int tid)
{
    for (int j = 0; j < n; ++j) {
        if (tid == 0) {
            float d = Sm[j * ld + j];
            Sm[j * ld + j] = sqrtf(fmaxf(d, 1e-30f));
        }
        __syncthreads();
        const float dj = Sm[j * ld + j];
        for (int i = j + 1 + tid; i < n; i += NTHR)
            Sm[i * ld + j] /= dj;
        __syncthreads();
        const int rem = n - 1 - j;
        for (int idx = tid; idx < rem * rem; idx += NTHR) {
            const int i  = j + 1 + idx / rem;
            const int k2 = j + 1 + idx % rem;
            if (k2 <= i)
                Sm[i * ld + k2] -= Sm[i * ld + j] * Sm[k2 * ld + j];
        }
        __syncthreads();
    }
}

__global__ __launch_bounds__(NTHR)
void kalman_smooth_kernel(const float* __restrict__ x,
                          const unsigned char* __restrict__ mask,
                          const float* __restrict__ Ain,
                          const float* __restrict__ Cin,
                          const float* __restrict__ muin,
                          const float* __restrict__ Sigin,
                          const float* __restrict__ Qin,
                          const float* __restrict__ Rin,
                          float* __restrict__ out_mu,
                          float* __restrict__ out_sig)
{
    __shared__ float sA [DZ_ * DZ_];   // A                           16 KB
    __shared__ float sC [DX_ * DZ_];   // C                           32 KB
    __shared__ float sP [DZ_ * DZ_];   // P_pred carry                16 KB
    __shared__ float sPf[DZ_ * DZ_];   // P_filt (current t)          16 KB
    __shared__ float sPs[DZ_ * DZ_];   // P_smooth carry              16 KB
    __shared__ float sW [DZ_ * DZ_];   // scratch: G=Z^T Z / AP / J   16 KB
    __shared__ float sCP[DX_ * DZ_];   // CP -> Z ; smoother: D | T1  32 KB
    __shared__ float sS [DX_ * DX_];   // S -> chol(S); P_pn          64 KB
    __shared__ float sQ2[DZ_], sR2[DX_];
    __shared__ float sMp[DZ_], sMf[DZ_], sMs[DZ_], sMpn[DZ_], sDm[DZ_];
    __shared__ float sVin[DX_], sVw[DX_];

    const int tid = threadIdx.x;
    const int b   = blockIdx.x;

    // ---- load static operators + initial state -------------------------
    for (int i = tid; i < DZ_ * DZ_; i += NTHR) { sA[i] = Ain[i]; sP[i] = 0.f; }
    for (int i = tid; i < DX_ * DZ_; i += NTHR) sC[i] = Cin[i];
    __syncthreads();
    if (tid < DZ_) {
        const float s = Sigin[tid];
        sP[tid * DZ_ + tid] = s * s;           // P0 = diag(Sigma^2)
        sMp[tid] = muin[tid];
        const float q = Qin[tid];
        sQ2[tid] = q * q;
    }
    if (tid < DX_) { const float rr = Rin[tid]; sR2[tid] = rr * rr; }
    __syncthreads();

    // =================== forward Kalman filter =========================
    for (int t = 0; t < T_; ++t) {
        // CP = C * P_pred                                  (128x64x64)
        wg_gemm<DX_, DZ_, DZ_, false, false, false>(sC, DZ_, sP, DZ_, sCP, DZ_, nullptr, 0, tid);
        __syncthreads();
        // S = CP * C^T                                     (128x128x64)
        wg_gemm<DX_, DX_, DZ_, false, true, false>(sCP, DZ_, sC, DZ_, sS, DX_, nullptr, 0, tid);
        __syncthreads();
        if (tid < DX_) {
            sS[tid * DX_ + tid] += sR2[tid];                // + diag(R^2)
            float a = x[((size_t)b * T_ + t) * DX_ + tid];  // innov = x - C m
            for (int k = 0; k < DZ_; ++k) a -= sC[tid * DZ_ + k] * sMp[k];
            sVin[tid] = a;
        }
        __syncthreads();
        wg_cholesky(sS, DX_, DX_, tid);                     // S = L L^T
        // Z = L^-1 CP  (column-parallel, in place) ; w = L^-1 innov
        if (tid < DZ_) {
            for (int i = 0; i < DX_; ++i) {
                float s = sCP[i * DZ_ + tid];
                for (int k = 0; k < i; ++k) s -= sS[i * DX_ + k] * sCP[k * DZ_ + tid];
                sCP[i * DZ_ + tid] = s / sS[i * DX_ + i];
            }
        } else if (tid == DZ_) {
            for (int i = 0; i < DX_; ++i) {
                float s = sVin[i];
                for (int k = 0; k < i; ++k) s -= sS[i * DX_ + k] * sVw[k];
                sVw[i] = s / sS[i * DX_ + i];
            }
        }
        __syncthreads();
        // G = Z^T Z                                        (64x64x128)
        wg_gemm<DZ_, DZ_, DX_, true, false, false>(sCP, DZ_, sCP, DZ_, sW, DZ_, nullptr, 0, tid);
        __syncthreads();
        const bool miss = (mask[(size_t)b * T_ + t] != 0);
        if (tid < DZ_) {
            float mm = sMp[tid];
            if (!miss) {
                float a = 0.f;
                for (int i = 0; i < DX_; ++i) a += sCP[i * DZ_ + tid] * sVw[i];
                mm += a;                                    // m_f = m + Z^T w
            }
            sMf[tid] = mm;
        }
        for (int idx = tid; idx < DZ_ * DZ_; idx += NTHR) { // P_f = sym(P - G)
            const int i2 = idx >> 6, j2 = idx & 63;
            sPf[idx] = miss ? sP[idx]
                            : 0.5f * ((sP[idx] - sW[idx]) +
                                      (sP[j2 * DZ_ + i2] - sW[j2 * DZ_ + i2]));
        }
        __syncthreads();
        // stage filter results in d_out (smoother overwrites in place)
        if (tid < DZ_) out_mu[((size_t)b * T_ + t) * DZ_ + tid] = sMf[tid];
        for (int idx = tid; idx < DZ_ * DZ_; idx += NTHR)
            out_sig[((size_t)b * T_ + t) * DZ_ * DZ_ + idx] = sPf[idx];
        // predict: P <- A P_f A^T + Q^2 ; m <- A m_f
        wg_gemm<DZ_, DZ_, DZ_, false, false, false>(sA, DZ_, sPf, DZ_, sW, DZ_, nullptr, 0, tid);
        __syncthreads();
        wg_gemm<DZ_, DZ_, DZ_, false, true, false>(sW, DZ_, sA, DZ_, sP, DZ_, nullptr, 0, tid);
        if (tid < DZ_) {
            float a = 0.f;
            for (int k = 0; k < DZ_; ++k) a += sA[tid * DZ_ + k] * sMf[k];
            sMpn[tid] = a;
        }
        __syncthreads();
        if (tid < DZ_) { sP[tid * DZ_ + tid] += sQ2[tid]; sMp[tid] = sMpn[tid]; }
        __syncthreads();
    }

    // =================== backward RTS smoother =========================
    for (int idx = tid; idx < DZ_ * DZ_; idx += NTHR) sPs[idx] = sPf[idx];
    if (tid < DZ_) sMs[tid] = sMf[tid];
    __syncthreads();

    float* const sD  = sCP;              // 64x64
    float* const sT1 = sCP + DZ_ * DZ_;  // 64x64

    for (int t = T_ - 2; t >= 0; --t) {
        // reload filtered (t)
        if (tid < DZ_) sMf[tid] = out_mu[((size_t)b * T_ + t) * DZ_ + tid];
        for (int idx = tid; idx < DZ_ * DZ_; idx += NTHR)
            sPf[idx] = out_sig[((size_t)b * T_ + t) * DZ_ * DZ_ + idx];
        __syncthreads();
        // AP = A P_f ; P_pn = AP A^T + Q^2 ; m_pn = A m_f
        wg_gemm<DZ_, DZ_, DZ_, false, false, false>(sA, DZ_, sPf, DZ_, sW, DZ_, nullptr, 0, tid);
        __syncthreads();
        wg_gemm<DZ_, DZ_, DZ_, false, true, false>(sW, DZ_, sA, DZ_, sS, DZ_, nullptr, 0, tid);
        if (tid < DZ_) {
            float a = 0.f;
            for (int k = 0; k < DZ_; ++k) a += sA[tid * DZ_ + k] * sMf[k];
            sMpn[tid] = a;
        }
        __syncthreads();
        if (tid < DZ_) {
            sS[tid * DZ_ + tid] += sQ2[tid];
            sDm[tid] = sMs[tid] - sMpn[tid];
        }
        __syncthreads();
        for (int idx = tid; idx < DZ_ * DZ_; idx += NTHR)   // D = P_s - P_pn
            sD[idx] = sPs[idx] - sS[idx];
        __syncthreads();
        wg_cholesky(sS, DZ_, DZ_, tid);                     // P_pn = L L^T
        // J = P_pn^-1 AP : forward then backward substitution, in place on sW
        if (tid < DZ_) {
            for (int i = 0; i < DZ_; ++i) {
                float s = sW[i * DZ_ + tid];
                for (int k = 0; k < i; ++k) s -= sS[i * DZ_ + k] * sW[k * DZ_ + tid];
                sW[i * DZ_ + tid] = s / sS[i * DZ_ + i];
            }
        }
        __syncthreads();
        if (tid < DZ_) {
            for (int i = DZ_ - 1; i >= 0; --i) {
                float s = sW[i * DZ_ + tid];
                for (int k = i + 1; k < DZ_; ++k) s -= sS[k * DZ_ + i] * sW[k * DZ_ + tid];
                sW[i * DZ_ + tid] = s / sS[i * DZ_ + i];
            }
        }
        __syncthreads();
        // T1 = J^T D ; P_new = T1 J + P_f   (into the carry sPs)
        wg_gemm<DZ_, DZ_, DZ_, true, false, false>(sW, DZ_, sD, DZ_, sT1, DZ_, nullptr, 0, tid);
        __syncthreads();
        wg_gemm<DZ_, DZ_, DZ_, false, false, true>(sT1, DZ_, sW, DZ_, sPs, DZ_, sPf, DZ_, tid);
        if (tid < DZ_) {                  // m_new = m_f + J^T (m_s - m_pn)
            float a = sMf[tid];
            for (int i = 0; i < DZ_; ++i) a += sW[i * DZ_ + tid] * sDm[i];
            sMs[tid] = a;
        }
        __syncthreads();
        // symmetrize P_new (two-phase to avoid in-place races)
        for (int idx = tid; idx < DZ_ * DZ_; idx += NTHR) sD[idx] = sPs[idx];
        __syncthreads();
        for (int idx = tid; idx < DZ_ * DZ_; idx += NTHR) {
            const int i2 = idx >> 6, j2 = idx & 63;
            sPs[idx] = 0.5f * (sD[idx] + sD[j2 * DZ_ + i2]);
        }
        __syncthreads();
        if (tid < DZ_) out_mu[((size_t)b * T_ + t) * DZ_ + tid] = sMs[tid];
        for (int idx = tid; idx < DZ_ * DZ_; idx += NTHR)
            out_sig[((size_t)b * T_ + t) * DZ_ * DZ_ + idx] = sPs[idx];
        __syncthreads();
    }
}

extern "C" void kernel_launch(void* const* d_in, const int* in_sizes, int n_in,
                              void* d_out, int out_size, void* d_ws, size_t ws_size,
                              hipStream_t stream)
{
    (void)in_sizes; (void)n_in; (void)out_size; (void)d_ws; (void)ws_size;
    const float*         x    = (const float*)d_in[0];
    const unsigned char* mask = (const unsigned char*)d_in[1];
    const float*         A    = (const float*)d_in[2];
    const float*         C    = (const float*)d_in[3];
    const float*         mu   = (const float*)d_in[4];
    const float*         Sig  = (const float*)d_in[5];
    const float*         Q    = (const float*)d_in[6];
    const float*         R    = (const float*)d_in[7];
    float* out_mu  = (float*)d_out;                         // [B,T,DZ]
    float* out_sig = out_mu + (size_t)B_ * T_ * DZ_;        // [B,T,DZ,DZ]
    kalman_smooth_kernel<<<dim3(B_), dim3(NTHR), 0, stream>>>(
        x, mask, A, C, mu, Sig, Q, R, out_mu, out_sig);
}